// TwoWordPSDProbe_71012989272563
// MI455X (gfx1250) — compile-verified
//
#include <hip/hip_runtime.h>
#include <hip/hip_bf16.h>

// Problem constants (match the JAX reference)
#define NPTS      2048
#define MODEL_DIM 1024
#define RANK      64

typedef float v2f __attribute__((ext_vector_type(2)));
typedef float v8f __attribute__((ext_vector_type(8)));

// ---------------------------------------------------------------------------
// Kernel 1: T = R @ P   (2048x1024 @ 1024x64 -> 2048x64, fp32 WMMA 16x16x4)
//
// One wave computes one 16x16 tile of T. 512 tiles total -> 64 blocks x 8 waves.
// A operand (16x4 f32): lane m in [0,16) holds K = {k0, k0+1} of row rI+m,
// lane m+16 holds K = {k0+2, k0+3} of the same row -> one float2 load per lane.
// B operand (4x16 f32): lane n holds column cB+n; VGPR0/1 = K pair selected by
// the lane's upper-half bit.
// ---------------------------------------------------------------------------
__global__ __launch_bounds__(256) void psd_project_kernel(
    const float* __restrict__ R, const float* __restrict__ P,
    float* __restrict__ T) {
  const int lane = threadIdx.x & 31;
  const int wave = threadIdx.x >> 5;
  const int tile = blockIdx.x * 8 + wave;          // 0..511
  const int rI = (tile >> 2) * 16;                 // row tile base (0..2032)
  const int cB = (tile & 3) * 16;                  // col tile base (0..48)

  const int m    = lane & 15;                      // row (A) / col (B) in tile
  const int koff = (lane >> 4) * 2;                // 0 for lanes 0-15, 2 for 16-31

  const float* aPtr = R + (size_t)(rI + m) * MODEL_DIM + koff;
  const float* bCol = P + cB + m;                  // stride RANK per K row

  v8f acc = {};
#pragma unroll 8
  for (int s = 0; s < MODEL_DIM / 4; ++s) {
    const int k0 = 4 * s;
    v2f a = *(const v2f*)(aPtr + k0);
    v2f b;
    b.x = bCol[(size_t)(k0 + koff) * RANK];
    b.y = bCol[(size_t)(k0 + koff + 1) * RANK];
    acc = __builtin_amdgcn_wmma_f32_16x16x4_f32(
        /*neg_a=*/false, a, /*neg_b=*/false, b,
        /*c_mod=*/(short)0, acc, /*reuse_a=*/false, /*reuse_b=*/false);
  }

  // C/D layout: VGPR r -> M = r (lanes 0-15) or r+8 (lanes 16-31), N = lane&15
  const int mBase = rI + ((lane >> 4) << 3);
#pragma unroll
  for (int r = 0; r < 8; ++r) {
    T[(size_t)(mBase + r) * RANK + cB + m] = acc[r];
  }
}

// ---------------------------------------------------------------------------
// Kernel 2: norms[i] = sum_r T[i][r]^2
// ---------------------------------------------------------------------------
__global__ __launch_bounds__(256) void psd_rownorm_kernel(
    const float* __restrict__ T, float* __restrict__ norms) {
  const int i = blockIdx.x * blockDim.x + threadIdx.x;
  if (i < NPTS) {
    const float4* row = (const float4*)(T + (size_t)i * RANK);
    float s = 0.0f;
#pragma unroll
    for (int q = 0; q < RANK / 4; ++q) {
      float4 v = row[q];
      s += v.x * v.x + v.y * v.y + v.z * v.z + v.w * v.w;
    }
    norms[i] = s;
  }
}

// ---------------------------------------------------------------------------
// Kernel 3: D[i,j] = norms[i] + norms[j] - 2 * (T T^T)[i,j]
//
// One wave per 16x16 output tile; 128x128 tiles -> 2048 blocks x 8 waves.
// A = T rows rI..rI+15, B = T^T (i.e. T rows rJ..rJ+15) -> identical float2
// per-lane fetch pattern for both operands. T (512 KB) stays L2-resident, so
// the kernel streams only the 16.8 MB output (the true bandwidth limit).
// ---------------------------------------------------------------------------
__global__ __launch_bounds__(256) void psd_gram_dist_kernel(
    const float* __restrict__ T, const float* __restrict__ norms,
    float* __restrict__ out) {
  const int lane = threadIdx.x & 31;
  const int wave = threadIdx.x >> 5;

  const int tileI = blockIdx.x >> 4;               // 0..127
  const int tileJ = (blockIdx.x & 15) * 8 + wave;  // 0..127
  const int rI = tileI * 16;
  const int rJ = tileJ * 16;

  const int m    = lane & 15;
  const int koff = (lane >> 4) * 2;

  const float* aPtr = T + (size_t)(rI + m) * RANK + koff;
  const float* bPtr = T + (size_t)(rJ + m) * RANK + koff;

  v8f acc = {};
#pragma unroll
  for (int s = 0; s < RANK / 4; ++s) {             // 16 WMMAs, K = 64
    v2f a = *(const v2f*)(aPtr + 4 * s);
    v2f b = *(const v2f*)(bPtr + 4 * s);
    acc = __builtin_amdgcn_wmma_f32_16x16x4_f32(
        false, a, false, b, (short)0, acc, false, false);
  }

  const float nj = norms[rJ + m];                  // N = lane&15 for this lane
  const int mBase = rI + ((lane >> 4) << 3);
#pragma unroll
  for (int r = 0; r < 8; ++r) {
    const int row = mBase + r;
    const float val = norms[row] + nj - 2.0f * acc[r];
    out[(size_t)row * NPTS + rJ + m] = val;
  }
}

// ---------------------------------------------------------------------------
// Host-side launcher
// ---------------------------------------------------------------------------
extern "C" void kernel_launch(void* const* d_in, const int* in_sizes, int n_in,
                              void* d_out, int out_size, void* d_ws, size_t ws_size,
                              hipStream_t stream) {
  const float* R = (const float*)d_in[0];   // (2048, 1024) fp32
  const float* P = (const float*)d_in[1];   // (1024, 64)   fp32
  float* out = (float*)d_out;               // (2048, 2048) fp32

  float* T     = (float*)d_ws;              // 2048*64 fp32 = 512 KB
  float* norms = T + (size_t)NPTS * RANK;   // 2048 fp32

  // T = R @ P : 512 tiles of 16x16, one wave per tile, 8 waves per block
  psd_project_kernel<<<64, 256, 0, stream>>>(R, P, T);

  // row norms
  psd_rownorm_kernel<<<NPTS / 256, 256, 0, stream>>>(T, norms);

  // pairwise squared distances: 128x128 tiles, 8 waves per block
  psd_gram_dist_kernel<<<2048, 256, 0, stream>>>(T, norms, out);
}